// SpatialGraphConvLayer_7490422964881
// MI455X (gfx1250) — compile-verified
//
#include <hip/hip_runtime.h>
#include <hip/hip_bf16.h>

// ---------------------------------------------------------------------------
// Problem constants (from reference setup_inputs)
//   x: [B=2, T=4, N=50000, C=32]  -> M = B*T = 8 slices
//   edge_index: [2, E=800000] int64, edge_weight: [E] f32
//   W: [K=3, C=32, O=32], bias/gamma/beta: [32]
// ---------------------------------------------------------------------------
#define NB_N   50000
#define NB_M   8
#define NB_C   32
#define NB_O   32
#define ROWS   (NB_N * NB_M)          // 400000 rows of the flattened feature mat
#define FEATS  (NB_M * NB_C)          // 256 floats per node
#define BN_EPS 1e-5f

typedef __attribute__((ext_vector_type(2))) float v2f;
typedef __attribute__((ext_vector_type(8))) float v8f;

__device__ __forceinline__ void atomAddF(float* p, float v) {
#if defined(__gfx1250__)
  unsafeAtomicAdd(p, v);        // hardware global_atomic_add_f32
#else
  atomicAdd(p, v);
#endif
}

// ---------------------------------------------------------------------------
// Generic zero fill
// ---------------------------------------------------------------------------
__global__ void fill_zero(float* __restrict__ p, int count) {
  int i = blockIdx.x * blockDim.x + threadIdx.x;
  if (i < count) p[i] = 0.0f;
}

// ---------------------------------------------------------------------------
// Degree: deg[src[e]] += w[e]
// ---------------------------------------------------------------------------
__global__ void deg_kernel(const long long* __restrict__ src,
                           const float* __restrict__ w,
                           float* __restrict__ deg, int E) {
  int e = blockIdx.x * blockDim.x + threadIdx.x;
  if (e < E) atomAddF(&deg[(int)src[e]], w[e]);
}

// deg -> d^{-1/2} in place
__global__ void dinv_kernel(float* __restrict__ deg, int N) {
  int n = blockIdx.x * blockDim.x + threadIdx.x;
  if (n < N) {
    float d = deg[n];
    deg[n] = (d > 0.0f) ? rsqrtf(d) : 0.0f;
  }
}

// norm[e] = -dinv[src]*w*dinv[dst]
__global__ void norm_kernel(const long long* __restrict__ src,
                            const long long* __restrict__ dst,
                            const float* __restrict__ w,
                            const float* __restrict__ dinv,
                            float* __restrict__ norm, int E) {
  int e = blockIdx.x * blockDim.x + threadIdx.x;
  if (e < E) norm[e] = -dinv[(int)src[e]] * w[e] * dinv[(int)dst[e]];
}

// ---------------------------------------------------------------------------
// x [M,N,C] -> T0 [N,M,C]
// ---------------------------------------------------------------------------
__global__ void transpose_x(const float* __restrict__ x,
                            float* __restrict__ T0, int total) {
  int tid = blockIdx.x * blockDim.x + threadIdx.x;
  if (tid >= total) return;
  int c = tid & 31;
  int m = (tid >> 5) & 7;
  int n = tid >> 8;
  T0[tid] = x[((size_t)m * NB_N + n) * NB_C + c];
}

// ---------------------------------------------------------------------------
// SpMM: zout[dst] += norm[e] * zin[src], rows of 256 floats.
// One wave per edge; float4 vectorized loads (2 x 512B coalesced per row).
// ---------------------------------------------------------------------------
__global__ __launch_bounds__(256) void spmm_kernel(
    const long long* __restrict__ src, const long long* __restrict__ dst,
    const float* __restrict__ norm, const float* __restrict__ zin,
    float* __restrict__ zout, int E) {
  long long gtid = (long long)blockIdx.x * blockDim.x + threadIdx.x;
  int e = (int)(gtid >> 5);
  if (e >= E) return;
  int lane = (int)(gtid & 31);
  int s = (int)src[e];
  int d = (int)dst[e];
  float w = norm[e];
  const float4* zs = (const float4*)(zin + (size_t)s * FEATS);
  float* zd = zout + (size_t)d * FEATS;
#pragma unroll
  for (int i = 0; i < 2; ++i) {
    int chunk = lane + i * 32;           // 64 float4 chunks per row
    float4 v = zs[chunk];
    float* q = zd + chunk * 4;
    atomAddF(q + 0, w * v.x);
    atomAddF(q + 1, w * v.y);
    atomAddF(q + 2, w * v.z);
    atomAddF(q + 3, w * v.w);
  }
}

// ---------------------------------------------------------------------------
// Effective weights folding T2 = 2*lap(T1) - T0:
//   out = T0@(W0-W2) + T1@W1 + C2@(2*W2)
// ---------------------------------------------------------------------------
__global__ void weff_kernel(const float* __restrict__ W,
                            float* __restrict__ Weff) {
  int t = blockIdx.x * blockDim.x + threadIdx.x;
  if (t < 1024) {
    Weff[t]        = W[t] - W[2048 + t];
    Weff[1024 + t] = W[1024 + t];
    Weff[2048 + t] = 2.0f * W[2048 + t];
  }
}

// ---------------------------------------------------------------------------
// Triple-term WMMA GEMM (fp32, V_WMMA_F32_16X16X4_F32).
// Block = 256 threads = 8 waves; each wave owns a 16x16 tile of OUT.
// Block covers 64 rows x 32 cols. ROWS = 400000 = 6250 * 64.
//
// A layout (16x4 f32): lane l holds A[l%16][4s + 2*(l/16) + {0,1}] in a.{x,y}
// B layout (4x16 f32): lane l holds B[4s + 2*(l/16) + {0,1}][l%16] in b.{x,y}
// C/D layout: VGPR r, lane l -> element (M = r + 8*(l/16), Ncol = l%16)
// ---------------------------------------------------------------------------
__global__ __launch_bounds__(256) void gemm_cheb(
    const float* __restrict__ T0, const float* __restrict__ T1,
    const float* __restrict__ C2, const float* __restrict__ Weff,
    const float* __restrict__ bias, float* __restrict__ outp) {
  const int lane    = threadIdx.x & 31;
  const int wave    = threadIdx.x >> 5;       // 0..7
  const int r0      = blockIdx.x * 64 + (wave >> 1) * 16;
  const int colBase = (wave & 1) * 16;
  const int mrow    = lane & 15;              // 0..15
  const int khalf   = (lane >> 4) * 2;        // 0 or 2

  v8f acc = {};
  const float* Aptr[3] = {T0, T1, C2};
#pragma unroll
  for (int kk = 0; kk < 3; ++kk) {
    const float* arow = Aptr[kk] + (size_t)(r0 + mrow) * NB_C;
    const float* B    = Weff + kk * 1024;
#pragma unroll
    for (int s = 0; s < 8; ++s) {
      const int kidx = s * 4 + khalf;
      v2f a, b;
      a.x = arow[kidx];
      a.y = arow[kidx + 1];
      b.x = B[kidx * NB_O + colBase + mrow];
      b.y = B[(kidx + 1) * NB_O + colBase + mrow];
      acc = __builtin_amdgcn_wmma_f32_16x16x4_f32(
          /*neg_a=*/false, a, /*neg_b=*/false, b,
          /*c_mod=*/(short)0, acc, /*reuse_a=*/false, /*reuse_b=*/false);
    }
  }
  const float bv   = bias[colBase + mrow];
  const int  rBase = r0 + 8 * (lane >> 4);
#pragma unroll
  for (int r = 0; r < 8; ++r)
    outp[(size_t)(rBase + r) * NB_O + colBase + mrow] = acc[r] + bv;
}

// ---------------------------------------------------------------------------
// BN partial sums: coalesced — thread t owns column t (pair = m*32+o),
// block strides over row chunks; combine partials with float atomics.
// ---------------------------------------------------------------------------
__global__ __launch_bounds__(256) void bn_partial(
    const float* __restrict__ outp, float* __restrict__ gsum,
    float* __restrict__ gsq, int rowsPerBlock) {
  const int pair = threadIdx.x;                  // 0..255
  int nBeg = blockIdx.x * rowsPerBlock;
  int nEnd = nBeg + rowsPerBlock;
  if (nEnd > NB_N) nEnd = NB_N;
  float sum = 0.0f, sq = 0.0f;
  for (int n = nBeg; n < nEnd; ++n) {
    float v = outp[(size_t)n * FEATS + pair];
    sum += v;
    sq  += v * v;
  }
  atomAddF(&gsum[pair], sum);
  atomAddF(&gsq[pair], sq);
}

__global__ void bn_finalize(const float* __restrict__ gsum,
                            const float* __restrict__ gsq,
                            const float* __restrict__ gamma,
                            const float* __restrict__ beta,
                            float* __restrict__ scale,
                            float* __restrict__ shift) {
  int p = threadIdx.x;                           // 256 threads, 1 block
  int o = p & 31;
  const float invN = 1.0f / (float)NB_N;
  float mean = gsum[p] * invN;
  float var  = gsq[p] * invN - mean * mean;
  float sc   = gamma[o] * rsqrtf(var + BN_EPS);
  scale[p] = sc;
  shift[p] = beta[o] - mean * sc;
}

// ---------------------------------------------------------------------------
// Epilogue: y[m,n,o] = relu(scale*pre[n,m,o] + shift)  (writes [B,T,N,O])
// ---------------------------------------------------------------------------
__global__ void bn_apply(const float* __restrict__ outp,
                         const float* __restrict__ scale,
                         const float* __restrict__ shift,
                         float* __restrict__ y, int total) {
  int tid = blockIdx.x * blockDim.x + threadIdx.x;
  if (tid >= total) return;
  int o = tid & 31;
  int n = (tid >> 5) % NB_N;
  int m = tid / (32 * NB_N);
  int pair = m * 32 + o;
  float v = outp[(size_t)n * FEATS + pair];
  v = scale[pair] * v + shift[pair];
  y[tid] = v > 0.0f ? v : 0.0f;
}

// ---------------------------------------------------------------------------
extern "C" void kernel_launch(void* const* d_in, const int* in_sizes, int n_in,
                              void* d_out, int out_size, void* d_ws, size_t ws_size,
                              hipStream_t stream) {
  const float*     x    = (const float*)d_in[0];
  const long long* ei   = (const long long*)d_in[1];   // int64 [2,E]
  const float*     ew   = (const float*)d_in[2];
  const float*     W    = (const float*)d_in[3];       // [3,32,32]
  const float*     bias = (const float*)d_in[4];
  const float*     gam  = (const float*)d_in[5];
  const float*     bet  = (const float*)d_in[6];

  const int E = in_sizes[2];                 // edge_weight count
  const long long* src = ei;
  const long long* dst = ei + E;

  const int total = ROWS * NB_C;             // 12,800,000

  // Workspace layout (floats)
  float* T0    = (float*)d_ws;               // [N,M,C]
  float* C2    = T0 + total;                 // lap(T1)
  float* OUT   = C2 + total;                 // pre-BN activations [N,M,O]
  float* deg   = OUT + total;                // N (deg -> dinv in place)
  float* norm  = deg + NB_N;                 // E
  float* Weff  = norm + E;                   // 3*1024
  float* gsum  = Weff + 3072;                // 256
  float* gsq   = gsum + 256;                 // 256
  float* scale = gsq + 256;                  // 256
  float* shift = scale + 256;                // 256

  float* T1 = (float*)d_out;                 // d_out doubles as T1 scratch

  const int TB = 256;
  const int gN     = (NB_N + TB - 1) / TB;
  const int gE     = (E + TB - 1) / TB;
  const int gFeat  = (total + TB - 1) / TB;
  const int gSpmm  = (int)(((long long)E * 32 + TB - 1) / TB);

  // 1) degrees and normalized edge weights
  fill_zero<<<gN, TB, 0, stream>>>(deg, NB_N);
  deg_kernel<<<gE, TB, 0, stream>>>(src, ew, deg, E);
  dinv_kernel<<<gN, TB, 0, stream>>>(deg, NB_N);
  norm_kernel<<<gE, TB, 0, stream>>>(src, dst, ew, deg, norm, E);

  // 2) T0 = x transposed to [N,M,C]
  transpose_x<<<gFeat, TB, 0, stream>>>(x, T0, total);

  // 3) T1 = lap(T0)   (accumulated into d_out scratch)
  fill_zero<<<gFeat, TB, 0, stream>>>(T1, total);
  spmm_kernel<<<gSpmm, TB, 0, stream>>>(src, dst, norm, T0, T1, E);

  // 4) C2 = lap(T1)
  fill_zero<<<gFeat, TB, 0, stream>>>(C2, total);
  spmm_kernel<<<gSpmm, TB, 0, stream>>>(src, dst, norm, T1, C2, E);

  // 5) effective weights + fused triple WMMA GEMM: OUT = bias + sum Tk @ Wk
  weff_kernel<<<4, TB, 0, stream>>>(W, Weff);
  gemm_cheb<<<ROWS / 64, TB, 0, stream>>>(T0, T1, C2, Weff, bias, OUT);

  // 6) BatchNorm statistics (biased variance over nodes, per (m,o))
  fill_zero<<<2, TB, 0, stream>>>(gsum, 512);
  const int nChunks = 512;
  const int rowsPerBlock = (NB_N + nChunks - 1) / nChunks;
  bn_partial<<<nChunks, TB, 0, stream>>>(OUT, gsum, gsq, rowsPerBlock);
  bn_finalize<<<1, TB, 0, stream>>>(gsum, gsq, gam, bet, scale, shift);

  // 7) normalize + ReLU + transpose to [B,T,N,O] (overwrites T1 scratch)
  bn_apply<<<gFeat, TB, 0, stream>>>(OUT, scale, shift, (float*)d_out, total);
}